// CCepLTVFilter_30837865185393
// MI455X (gfx1250) — compile-verified
//
#include <hip/hip_runtime.h>
#include <hip/hip_bf16.h>

// Problem constants (from reference)
#define BB    2
#define TT    128
#define DD    80
#define CCEP  222
#define FFTN  1024
#define HOP   256
#define WIN   512          // 2*HOP
#define PADQ  401          // (FFT - CCEP)/2
#define NIDFT 1025         // FFT+1
#define KRED  240          // D*3

typedef __attribute__((ext_vector_type(2))) float v2f;
typedef __attribute__((ext_vector_type(8))) float v8f;

// Fragment index helpers for V_WMMA_F32_16X16X4_F32 (wave32):
//   A 16x4 : lane l holds A[M=l&15, K=(l>>4)*2 + v], v in {0,1}
//   B 4x16 : lane l holds B[K=(l>>4)*2 + v, N=l&15]
//   C/D    : VGPR r: lanes 0-15 -> M=r, lanes 16-31 -> M=r+8, N=lane&15

// ---------------------------------------------------------------------------
// K1: conv_feat as fp32 WMMA GEMM.
//   ccep[bt, c] = sum_{kk,d} x[b, t+kk-1, d] * W[c, d, kk] + bias[c], then /quef.
// One wave per 16x16 tile; M = B*T = 256, N = 224 (c >= 222 masked), K = 240.
// ---------------------------------------------------------------------------
__global__ void k_conv_wmma(const float* __restrict__ x,
                            const float* __restrict__ W,
                            const float* __restrict__ bias,
                            float* __restrict__ ccep) {
    const int lane  = threadIdx.x & 31;
    const int half  = lane >> 4;
    const int l16   = lane & 15;
    const int tileN = blockIdx.x;          // 0..13
    const int tileM = blockIdx.y;          // 0..15

    const int m = tileM * 16 + l16;        // global row (bt) for A loads
    const int b = m >> 7;
    const int t = m & 127;
    const int n = tileN * 16 + l16;        // cepstral channel for B loads

    __builtin_prefetch(&W[(tileN * 16) * KRED], 0, 3);  // global_prefetch_b8

    v8f acc = {};
    for (int kq0 = 0; kq0 < KRED; kq0 += 4) {
        const int kq = kq0 + half * 2;
        v2f av, bv;
#pragma unroll
        for (int v = 0; v < 2; ++v) {
            const int kqe = kq + v;
            const int kk  = kqe / DD;          // conv tap 0..2
            const int d   = kqe - kk * DD;     // input channel
            const int ts  = t + kk - 1;
            av[v] = ((unsigned)ts < (unsigned)TT)
                        ? x[(b * TT + ts) * DD + d] : 0.0f;
            bv[v] = (n < CCEP) ? W[n * KRED + d * 3 + kk] : 0.0f;
        }
        acc = __builtin_amdgcn_wmma_f32_16x16x4_f32(
            false, av, false, bv, (short)0, acc, false, false);
    }

    const int c = n;
    if (c < CCEP) {
        const float qn  = (c < CCEP / 2) ? (float)(CCEP / 2 - c)
                                         : (float)(c - CCEP / 2 + 1);
        const float bia = bias[c];
        const float inv = 1.0f / qn;
#pragma unroll
        for (int r = 0; r < 8; ++r) {
            const int mrow = tileM * 16 + half * 8 + r;
            ccep[mrow * CCEP + c] = (acc[r] + bia) * inv;
        }
    }
}

// ---------------------------------------------------------------------------
// K2a: 1024-pt DFT of padded ccep rows as a shared-B WMMA GEMM, fused with
//      the pointwise c2 = 10^(Re/10) * cis(Im) map.
//   Yr[bt,k] = sum_m ccep[bt,m] * cos(2pi k (m+PADQ)/1024)
//   Yi[bt,k] = sum_m ccep[bt,m] * (-sin(2pi k (m+PADQ)/1024))
// M = 256 rows (bt), N = 1024 cols (k), K = 222 padded to 224.
// Twiddles come from a block-shared LDS table e^{-2pi i j/1024},
// index j = (k*(m+PADQ)) & 1023.  8 waves/block, one 16x16 tile each.
// ---------------------------------------------------------------------------
__global__ void k_dft_wmma(const float* __restrict__ ccep,
                           float* __restrict__ c2r,
                           float* __restrict__ c2i) {
    __shared__ float tc[FFTN];
    __shared__ float ts[FFTN];
    const int tid   = threadIdx.x;
    const int wid   = tid >> 5;
    const int lane  = tid & 31;
    const int half  = lane >> 4;
    const int l16   = lane & 15;
    const int tileN = blockIdx.x;                 // 0..63  (k tile)
    const int tileM = blockIdx.y * 8 + wid;       // 0..15  (bt tile)

    const float wang = -6.283185307179586f / (float)FFTN;
    for (int j = tid; j < FFTN; j += 256) {
        float s, c; __sincosf(wang * (float)j, &s, &c);
        tc[j] = c; ts[j] = s;
    }
    __syncthreads();

    const int bt = tileM * 16 + l16;              // A row
    const int k  = tileN * 16 + l16;              // B col (frequency)

    v8f accR = {}, accI = {};
    for (int kq0 = 0; kq0 < 224; kq0 += 4) {      // K = 222 padded to 224
        v2f av, bc, bs;
#pragma unroll
        for (int v = 0; v < 2; ++v) {
            const int mm = kq0 + half * 2 + v;    // quefrency index
            const bool ok = (mm < CCEP);
            av[v] = ok ? ccep[bt * CCEP + mm] : 0.0f;
            const int j = (k * (mm + PADQ)) & (FFTN - 1);
            bc[v] = ok ? tc[j] : 0.0f;
            bs[v] = ok ? ts[j] : 0.0f;
        }
        accR = __builtin_amdgcn_wmma_f32_16x16x4_f32(
            false, av, false, bc, (short)0, accR, false, false);
        accI = __builtin_amdgcn_wmma_f32_16x16x4_f32(
            false, av, false, bs, (short)0, accI, false, false);
    }

    // Pointwise: mag = 10^(Yr/10); c2 = mag * cis(Yi)
#pragma unroll
    for (int r = 0; r < 8; ++r) {
        const int row = tileM * 16 + half * 8 + r;
        const float mag = __expf(accR[r] * 0.23025850929940457f);
        float s, c; __sincosf(accI[r], &s, &c);
        c2r[row * FFTN + k] = mag * c;
        c2i[row * FFTN + k] = mag * s;
    }
}

// ---------------------------------------------------------------------------
// K2b: 1025-pt inverse DFT (real part) as a shared-B WMMA GEMM.
//   y[bt,n] = sum_k cr[bt,k]*cos(2pi k n/1025)/1025
//           + sum_k ci[bt,k]*(-sin(2pi k n/1025))/1025
// M = 256 (bt), N = 1025 -> 65 tiles (n >= 1025 masked), K = 1024.
// LDS table holds e^{+2pi i j/1025}/1025; per-lane index (k*n) mod 1025 is
// maintained incrementally: step = (4n) mod 1025, one conditional subtract.
// ---------------------------------------------------------------------------
__global__ void k_idft_wmma(const float* __restrict__ c2r,
                            const float* __restrict__ c2i,
                            float* __restrict__ y) {
    __shared__ float twc[NIDFT];
    __shared__ float tws[NIDFT];
    const int tid   = threadIdx.x;
    const int wid   = tid >> 5;
    const int lane  = tid & 31;
    const int half  = lane >> 4;
    const int l16   = lane & 15;
    const int tileN = blockIdx.x;                 // 0..64  (n tile)
    const int tileM = blockIdx.y * 8 + wid;       // 0..15  (bt tile)

    const float wang = 6.283185307179586f / (float)NIDFT;
    const float inv  = 1.0f / (float)NIDFT;
    for (int j = tid; j < NIDFT; j += 256) {
        float s, c; __sincosf(wang * (float)j, &s, &c);
        twc[j] = c * inv;
        tws[j] = -s * inv;
    }
    __syncthreads();

    const int bt   = tileM * 16 + l16;            // A row
    const int n    = tileN * 16 + l16;            // B col (output sample)
    const bool okn = (n < NIDFT);

    // Incremental (k*n) mod 1025 for this lane's two B elements.
    int j0   = ((half * 2 + 0) * n) % NIDFT;
    int j1   = ((half * 2 + 1) * n) % NIDFT;
    int step = (4 * n) % NIDFT;

    v8f acc = {};
    for (int kq0 = 0; kq0 < FFTN; kq0 += 4) {     // 256 steps
        const int kk = kq0 + half * 2;
        v2f ar, ai, bc, bs;
        ar[0] = c2r[bt * FFTN + kk];
        ar[1] = c2r[bt * FFTN + kk + 1];
        ai[0] = c2i[bt * FFTN + kk];
        ai[1] = c2i[bt * FFTN + kk + 1];
        bc[0] = okn ? twc[j0] : 0.0f;
        bc[1] = okn ? twc[j1] : 0.0f;
        bs[0] = okn ? tws[j0] : 0.0f;
        bs[1] = okn ? tws[j1] : 0.0f;
        j0 += step; if (j0 >= NIDFT) j0 -= NIDFT;
        j1 += step; if (j1 >= NIDFT) j1 -= NIDFT;

        acc = __builtin_amdgcn_wmma_f32_16x16x4_f32(
            false, ar, false, bc, (short)0, acc, false, false);
        acc = __builtin_amdgcn_wmma_f32_16x16x4_f32(
            false, ai, false, bs, (short)0, acc, false, false);
    }

    if (okn) {
#pragma unroll
        for (int r = 0; r < 8; ++r) {
            const int row = tileM * 16 + half * 8 + r;
            y[row * NIDFT + n] = acc[r];
        }
    }
}

// ---------------------------------------------------------------------------
// K3: conv_impulse collapsed to a per-(b,t) Toeplitz matvec (frame zero-pad
// leaves lags y[u + 512 - w], u in [0,512)).  Hann window fused.
// ---------------------------------------------------------------------------
__global__ void k_filt(const float* __restrict__ z,
                       const float* __restrict__ y,
                       float* __restrict__ zfw) {
    __shared__ float fr[WIN];
    __shared__ float yr[NIDFT];
    const int bt  = blockIdx.x;
    const int b   = bt >> 7;
    const int t   = bt & 127;
    const int tid = threadIdx.x;

    for (int u = tid; u < WIN; u += 256) {
        const int i = t * HOP + u;                  // index into left-padded zs
        fr[u] = (i >= HOP) ? z[b * (TT * HOP) + i - HOP] : 0.0f;
    }
    for (int k = tid; k < NIDFT; k += 256) yr[k] = y[bt * NIDFT + k];
    __syncthreads();

    for (int w = tid; w < WIN; w += 256) {
        float acc = 0.0f;
        const float* yp = &yr[WIN - w];
        for (int u = 0; u < WIN; ++u) acc = fmaf(fr[u], yp[u], acc);
        const float win =
            0.5f * (1.0f - __cosf(6.283185307179586f * (float)w / (float)WIN));
        zfw[bt * WIN + w] = acc * win;
    }
}

// ---------------------------------------------------------------------------
// K4: overlap-add.  out[b, t*HOP + h] = zfw[b,t,h] + zfw[b,(t-1)%T, h+HOP]
// ---------------------------------------------------------------------------
__global__ void k_ola(const float* __restrict__ zfw, float* __restrict__ out) {
    const int idx = blockIdx.x * 256 + threadIdx.x;   // 0..65535
    const int b = idx >> 15;
    const int r = idx & 32767;
    const int t = r >> 8;
    const int h = r & 255;
    const int tp = (t + TT - 1) & (TT - 1);
    out[idx] = zfw[(b * TT + t) * WIN + h] + zfw[(b * TT + tp) * WIN + h + HOP];
}

// ---------------------------------------------------------------------------
extern "C" void kernel_launch(void* const* d_in, const int* in_sizes, int n_in,
                              void* d_out, int out_size, void* d_ws, size_t ws_size,
                              hipStream_t stream) {
    const float* x    = (const float*)d_in[0];   // (B,T,D)
    const float* z    = (const float*)d_in[1];   // (B,1,T*HOP)
    const float* W    = (const float*)d_in[2];   // (CCEP,D,3)
    const float* bias = (const float*)d_in[3];   // (CCEP,)
    float* out = (float*)d_out;                  // (B,1,T*HOP)

    float* ws   = (float*)d_ws;
    float* ccep = ws;                    // 256*222  = 56832 (+pad)
    float* c2r  = ws + 57344;            // 256*1024 = 262144
    float* c2i  = ws + 319488;           // 256*1024 = 262144
    float* yimp = ws + 581632;           // 256*1025 = 262400
    float* zfw  = ws + 844032;           // 256*512  = 131072

    k_conv_wmma<<<dim3(14, 16),  32, 0, stream>>>(x, W, bias, ccep);
    k_dft_wmma <<<dim3(64, 2),  256, 0, stream>>>(ccep, c2r, c2i);
    k_idft_wmma<<<dim3(65, 2),  256, 0, stream>>>(c2r, c2i, yimp);
    k_filt     <<<BB * TT,      256, 0, stream>>>(z, yimp, zfw);
    k_ola      <<<(BB * TT * HOP) / 256, 256, 0, stream>>>(zfw, out);
}